// NCA_MMD_Alignment_7954279432506
// MI455X (gfx1250) — compile-verified
//
#include <hip/hip_runtime.h>

typedef __attribute__((ext_vector_type(2))) float v2f;
typedef __attribute__((ext_vector_type(8))) float v8f;

#define NPTS 4096
#define NTOT 8192
#define EPSF 1e-6f
#define MUF 18.0f
#define SIGMAF 8.0f
#define LOG2E 1.4426950408889634f

// workspace layout (float offsets)
#define OFF_SUMS  0          // [0]=sumS raw, [1]=sumT raw, [2]=c4 coeff, [3]=pad
#define OFF_WS    4          // ws_raw[4096]
#define OFF_WT    (4+4096)   // wt_raw[4096]
#define OFF_SQS   (4+8192)   // |s_i|^2 [4096]
#define OFF_SQT   (4+12288)  // |t_j|^2 [4096]
#define OFF_PART  (4+16384)  // block partials [1024]

// ---------------------------------------------------------------------------
// Kernel 1: per-point squared norms + raw confidence weights + scalar sums
// for bandwidth (sum(L2) = 2n*sum|x|^2 - 2*||sum x||^2) and weight norms.
// Stores the final exp2 coefficient c4 = -log2(e)/(16*bw) so kernel 2 needs
// no division. Single 1024-thread block; fixed-order LDS tree reduction.
// ---------------------------------------------------------------------------
__global__ void nca_prep_kernel(const float* __restrict__ src,
                                const float* __restrict__ tgt,
                                float* __restrict__ ws) {
  __shared__ float red[1024];
  const int t = threadIdx.x;
  float sumS = 0.f, sumT = 0.f, sumSq = 0.f, sx = 0.f, sy = 0.f, sz = 0.f;

  for (int p = t; p < NTOT; p += 1024) {
    const bool is_src = (p < NPTS);            // wave-uniform per iteration
    const float* base = is_src ? (src + 3 * p) : (tgt + 3 * (p - NPTS));
    const float x = base[0], y = base[1], z = base[2];
    const float sq = x * x + y * y + z * z;
    // phi = 2v + alpha + 1/(beta+eps); w = exp(-(phi-mu)^2 / (2 sigma^2))
    const float phi = 2.f * x + y + 1.f / (z + EPSF);
    const float dmu = phi - MUF;
    const float w = __builtin_amdgcn_exp2f(
        -dmu * dmu * (LOG2E / (2.f * SIGMAF * SIGMAF)));
    if (is_src) {
      ws[OFF_SQS + p] = sq;
      ws[OFF_WS + p]  = w;
      sumS += w;
    } else {
      const int q = p - NPTS;
      ws[OFF_SQT + q] = sq;
      ws[OFF_WT + q]  = w;
      sumT += w;
    }
    sumSq += sq; sx += x; sy += y; sz += z;
  }

  float vals[6] = {sumS, sumT, sumSq, sx, sy, sz};
  float out[6];
  for (int k = 0; k < 6; ++k) {
    red[t] = vals[k];
    __syncthreads();
    for (int s = 512; s > 0; s >>= 1) {
      if (t < s) red[t] += red[t + s];
      __syncthreads();
    }
    out[k] = red[0];
    __syncthreads();
  }

  if (t == 0) {
    const float n = (float)NTOT;
    const float sumL2 = 2.f * n * out[2]
                      - 2.f * (out[3] * out[3] + out[4] * out[4] + out[5] * out[5]);
    float bw = sumL2 / (n * n - n);
    bw *= 0.25f;  // / KERNEL_MUL^(KERNEL_NUM//2) = / 2^2
    ws[0] = out[0];
    ws[1] = out[1];
    ws[2] = -LOG2E / (16.f * bw);  // c4: smallest-scale exp2 coefficient
  }
}

// ---------------------------------------------------------------------------
// Kernel 2: 128x128 macro-tile per 256-thread block (8 waves, 1 tile-row each).
// Per 16x16 tile: dot(s_i, t_j) via V_WMMA_F32_16X16X4_F32 (K=4, zero-padded).
// Multiscale kernel sum uses ONE v_exp_f32 per element:
//   u = exp2(c4*L2)  =>  sum_i exp2(c_i*L2) = u + u^2 + u^4 + u^8 + u^16
// (c_i = c0/2^i, so larger scales are repeated squarings of the smallest).
// Branch-free fragment construction (unconditional float3 loads + cndmask)
// keeps EXEC = all-ones. Fixed-order LDS reduce -> one partial per block.
// ---------------------------------------------------------------------------
__global__ void nca_mmd_tile_kernel(const float* __restrict__ src,
                                    const float* __restrict__ tgt,
                                    const float* __restrict__ wsbuf,
                                    float* __restrict__ partials) {
  const float* __restrict__ ws_raw = wsbuf + OFF_WS;
  const float* __restrict__ wt_raw = wsbuf + OFF_WT;
  const float* __restrict__ sq_s   = wsbuf + OFF_SQS;
  const float* __restrict__ sq_t   = wsbuf + OFF_SQT;
  const float3* __restrict__ src3  = reinterpret_cast<const float3*>(src);
  const float3* __restrict__ tgt3  = reinterpret_cast<const float3*>(tgt);

  const float c4 = wsbuf[2];                  // precomputed: -log2e/(16*bw)

  const int lane = threadIdx.x & 31;
  const int wave = threadIdx.x >> 5;
  const int lmod = lane & 15;
  const int hi   = lane >> 4;                 // 0: K0/K1 half, 1: K2/K3 half
  const int mBase = blockIdx.y * 128 + wave * 16;
  const int nBase = blockIdx.x * 128;

  // A fragment: 16x4 f32. lanes 0-15 hold {x,y} (K0,K1); lanes 16-31 {z,0}.
  const float3 sp = src3[mBase + lmod];
  v2f a;
  a.x = hi ? sp.z : sp.x;
  a.y = hi ? 0.f  : sp.y;

  // Row-constant data (fixed for this wave across all column tiles).
  float sqs_r[8], wsr[8];
#pragma unroll
  for (int r = 0; r < 8; ++r) {
    const int M = mBase + 8 * hi + r;         // C/D layout: M = vgpr + 8*(lane/16)
    sqs_r[r] = sq_s[M];
    wsr[r]   = ws_raw[M];
  }

  float acc = 0.f;
#pragma unroll
  for (int j = 0; j < 8; ++j) {
    const int n = nBase + j * 16 + lmod;      // C/D layout: N = lane % 16
    // B fragment: 4x16 f32, mirrored layout of A.
    const float3 tp = tgt3[n];
    v2f b;
    b.x = hi ? tp.z : tp.x;
    b.y = hi ? 0.f  : tp.y;

    v8f d = {};
    d = __builtin_amdgcn_wmma_f32_16x16x4_f32(
        /*neg_a=*/false, a, /*neg_b=*/false, b,
        /*c_mod=*/(short)0, d, /*reuse_a=*/false, /*reuse_b=*/false);

    const float sqt = sq_t[n];
    const float wt  = wt_raw[n];
    float rowacc = 0.f;
#pragma unroll
    for (int r = 0; r < 8; ++r) {
      const float L2 = sqs_r[r] + sqt - 2.f * d[r];
      const float u   = __builtin_amdgcn_exp2f(c4 * L2);
      const float u2  = u * u;
      const float u4  = u2 * u2;
      const float u8  = u4 * u4;
      const float u16 = u8 * u8;
      const float k = ((u + u2) + (u4 + u8)) + u16;
      rowacc += k * wsr[r];
    }
    acc += rowacc * wt;
  }

  __shared__ float red[256];
  red[threadIdx.x] = acc;
  __syncthreads();
  for (int s = 128; s > 0; s >>= 1) {
    if (threadIdx.x < s) red[threadIdx.x] += red[threadIdx.x + s];
    __syncthreads();
  }
  if (threadIdx.x == 0)
    partials[blockIdx.y * gridDim.x + blockIdx.x] = red[0];
}

// ---------------------------------------------------------------------------
// Kernel 3: reduce 1024 block partials (fixed order) + final scaling.
// ---------------------------------------------------------------------------
__global__ void nca_finalize_kernel(const float* __restrict__ wsbuf,
                                    const float* __restrict__ partials,
                                    float* __restrict__ out) {
  __shared__ float red[256];
  const int t = threadIdx.x;
  float s = 0.f;
  for (int i = t; i < 1024; i += 256) s += partials[i];
  red[t] = s;
  __syncthreads();
  for (int st = 128; st > 0; st >>= 1) {
    if (t < st) red[t] += red[t + st];
    __syncthreads();
  }
  if (t == 0) {
    const float U = red[0];
    const float denom = (wsbuf[0] + EPSF) * (wsbuf[1] + EPSF);
    out[0] = -2.f * U / denom;
  }
}

extern "C" void kernel_launch(void* const* d_in, const int* in_sizes, int n_in,
                              void* d_out, int out_size, void* d_ws, size_t ws_size,
                              hipStream_t stream) {
  const float* src = (const float*)d_in[0];
  const float* tgt = (const float*)d_in[1];
  float* ws  = (float*)d_ws;
  float* out = (float*)d_out;

  nca_prep_kernel<<<1, 1024, 0, stream>>>(src, tgt, ws);

  dim3 grid(32, 32);  // 4096/128 in each dim
  nca_mmd_tile_kernel<<<grid, 256, 0, stream>>>(src, tgt, ws, ws + OFF_PART);

  nca_finalize_kernel<<<1, 256, 0, stream>>>(ws, ws + OFF_PART, out);
}